// RectUpsamplerWithOrog_51737176048340
// MI455X (gfx1250) — compile-verified
//
#include <hip/hip_runtime.h>
#include <hip/hip_bf16.h>

// RectUpsamplerWithOrog — MI455X (gfx1250, wave32) implementation.
//
// Per high-res point p, the 27-term gather-dot over (b,f) is cast as a fp32
// matmul  D(64x16) = A(64x56) x B(56x16)  via V_WMMA_F32_16X16X4_F32 (fp32
// in/out, so no precision loss vs the fp32 reference).  B columns enumerate
// (cls,f); row b consumes column cls_ids[b]*4+f.  K layout:
//   kk = f'*9+k  (0..35)  -> y_low[b,f',lowmap(nb[p,k])]   x  w[cls,f,p,k,0]*(f'==f)
//   kk = 36+2k+d (36..53) -> orog[b,nb,d]-bias_orog[cls,d,nb] x w[cls,f,p,k,1+d]
//   kk = 54,55            -> zero padding
// All conditional values are produced branchlessly (unconditional loads with
// v_cndmask-selected indices / zero masks) so the WMMA loop has no exec-mask
// traffic.  Low-res field (32KB) and bias_orog rows (20KB) staged in LDS.

typedef float v2f __attribute__((ext_vector_type(2)));
typedef float v8f __attribute__((ext_vector_type(8)));

#define PHI   16384   // 128*128
#define PLO   256     // 16*16
#define WSTR  27      // 9*(1+2) floats per (cls,f,p)

__global__ __launch_bounds__(256)
void rect_upsampler_orog_wmma(const float* __restrict__ x,          // (64, 4*256)
                              const float* __restrict__ orog,       // (64, 16384, 2)
                              const float* __restrict__ wm,         // (4,4,16384,9,3)
                              const float* __restrict__ bias_low,   // (4,4,256)
                              const float* __restrict__ bias_high,  // (4,4,16384)
                              const float* __restrict__ bias_orog,  // (4,2,16384)
                              const int*   __restrict__ cls_ids,    // (64,)
                              const int*   __restrict__ nbi,        // (16384, 9)
                              float*       __restrict__ out)        // (64,4,128,128)
{
    __shared__ float ylow_s[32 * 256];   // [cell][b*4+f], cell=whichRow*16+col
    __shared__ float borog_s[8 * 640];   // [(cls*2+d)][q - qbase], 5 rows x 128
    __shared__ int   cls_s[64];

    const int tid = threadIdx.x;
    const int i   = blockIdx.x;                            // high-res row
    const int rA  = ((i >= 2) ? (i - 2) : 0) >> 3;         // candidate low rows
    const int rB  = (((i + 2) > 127) ? 127 : (i + 2)) >> 3;
    const int r0  = (i >= 2) ? (i - 2) : 0;                // first staged hi row
    const int qbase = r0 * 128;

    if (tid < 64) cls_s[tid] = cls_ids[tid];
    __syncthreads();

    // Stage y_low = x - bias_low[cls] for both candidate low rows, all (b,f).
    for (int idx = tid; idx < 32 * 256; idx += 256) {
        const int cell = idx >> 8;
        const int rem  = idx & 255;
        const int b    = rem >> 2;
        const int f    = rem & 3;
        const int r    = (cell < 16) ? rA : rB;
        const int low  = r * 16 + (cell & 15);
        const int cb   = cls_s[b];
        ylow_s[idx] = x[b * 1024 + f * 256 + low]
                    - bias_low[(cb * 4 + f) * PLO + low];
    }
    // Stage bias_orog for hi-res rows r0..r0+4 (clamped), all (cls,d).
    #pragma unroll
    for (int c2 = 0; c2 < 8; ++c2) {
        for (int qq = tid; qq < 640; qq += 256) {
            int q = qbase + qq;
            q = (q > PHI - 1) ? (PHI - 1) : q;   // clamped rows are never read
            borog_s[c2 * 640 + qq] = bias_orog[c2 * PHI + q];
        }
    }
    __syncthreads();

    const int lane  = tid & 31;
    const int wave  = tid >> 5;
    const int mtile = wave & 3;     // which 16 batches
    const int phalf = wave >> 2;    // left/right half of the row
    const int hi    = lane >> 4;    // lane half -> K slots {0,1} vs {2,3}
    const int b     = mtile * 16 + (lane & 15);   // A row (batch)
    const int b4    = b * 4;
    const int clsb  = cls_s[b];
    const int nf    = lane & 3;     // B column feature
    const int ncls  = (lane >> 2) & 3;            // B column class

    const size_t bOfs = (size_t)b * (2 * PHI);    // orog batch offset
    const int    cOfs = clsb * 1280;              // borog_s (cls) offset

    // f'==f column masks (constant-indexed -> registers).
    float mf[4];
    #pragma unroll
    for (int f = 0; f < 4; ++f) mf[f] = (nf == f) ? 1.0f : 0.0f;

    // Per-lane extraction constants for the 8 accumulator rows.
    int   outBase[8]; int bhBase[8]; bool match[8];
    #pragma unroll
    for (int j = 0; j < 8; ++j) {
        const int bb = mtile * 16 + j + hi * 8;
        const int cb = cls_s[bb];
        match[j]   = (ncls == cb);
        outBase[j] = (bb * 4 + nf) * PHI;
        bhBase[j]  = (cb * 4 + nf) * PHI;
    }

    for (int t = 0; t < 64; ++t) {
        const int p = i * 128 + phalf * 64 + t;
        const int* nb = nbi + p * 9;
        const float* wcol = wm + (size_t)((ncls * 4 + nf) * PHI + p) * WSTR;
        if (t + 1 < 64) __builtin_prefetch(wcol + WSTR, 0, 0);

        // ---- per-point preloads (all constant-indexed register arrays) ----
        int nbq[9];
        #pragma unroll
        for (int k = 0; k < 9; ++k) nbq[k] = nb[k];
        int cellA[9];                        // LDS base addr per neighbor
        #pragma unroll
        for (int k = 0; k < 9; ++k) {
            const int q = nbq[k];
            const int cell = (((q >> 10) == rA) ? 0 : 16) + ((q >> 3) & 15);
            cellA[k] = cell * 256 + b4;
        }
        float w[27];
        #pragma unroll
        for (int j = 0; j < 27; ++j) w[j] = wcol[j];

        // ---- compile-time helpers (kk always a constant after unroll) ----
        auto lowAddr = [&](int kk) -> int {      // kk < 36
            const int fp = kk / 9, k = kk - fp * 9;
            return cellA[k] + fp;
        };
        auto bLow = [&](int kk) -> float {       // kk < 36
            const int fp = kk / 9, k = kk - fp * 9;
            return w[k * 3] * mf[fp];
        };
        auto orogIdx = [&](int kk) -> int {      // 36 <= kk < 54
            const int tt = kk - 36, k = tt >> 1, d = tt & 1;
            return nbq[k] * 2 + d;
        };
        auto borogIdx = [&](int kk) -> int {     // 36 <= kk < 54
            const int tt = kk - 36, k = tt >> 1, d = tt & 1;
            return d * 640 + (nbq[k] - qbase);
        };
        auto bOrog = [&](int kk) -> float {      // 36 <= kk < 54
            const int tt = kk - 36, k = tt >> 1, d = tt & 1;
            return w[k * 3 + 1 + d];
        };

        v8f acc = {};
        #pragma unroll
        for (int c = 0; c < 14; ++c) {
            float a0, a1, b0, b1;
            if (c < 9) {
                // low-res terms: select LDS address by lane half, single load
                const int ad0 = hi ? lowAddr(c * 4 + 2) : lowAddr(c * 4 + 0);
                const int ad1 = hi ? lowAddr(c * 4 + 3) : lowAddr(c * 4 + 1);
                a0 = ylow_s[ad0];
                a1 = ylow_s[ad1];
                b0 = hi ? bLow(c * 4 + 2) : bLow(c * 4 + 0);
                b1 = hi ? bLow(c * 4 + 3) : bLow(c * 4 + 1);
            } else if (c < 13) {
                // orog terms: global gather + LDS bias, index-selected
                const int o0 = hi ? orogIdx(c * 4 + 2) : orogIdx(c * 4 + 0);
                const int o1 = hi ? orogIdx(c * 4 + 3) : orogIdx(c * 4 + 1);
                const int g0 = hi ? borogIdx(c * 4 + 2) : borogIdx(c * 4 + 0);
                const int g1 = hi ? borogIdx(c * 4 + 3) : borogIdx(c * 4 + 1);
                a0 = orog[bOfs + o0] - borog_s[cOfs + g0];
                a1 = orog[bOfs + o1] - borog_s[cOfs + g1];
                b0 = hi ? bOrog(c * 4 + 2) : bOrog(c * 4 + 0);
                b1 = hi ? bOrog(c * 4 + 3) : bOrog(c * 4 + 1);
            } else {
                // c == 13: lanes hi=0 carry kk=52,53; hi=1 lanes are padding.
                const float av0 = orog[bOfs + orogIdx(52)] - borog_s[cOfs + borogIdx(52)];
                const float av1 = orog[bOfs + orogIdx(53)] - borog_s[cOfs + borogIdx(53)];
                a0 = hi ? 0.0f : av0;
                a1 = hi ? 0.0f : av1;
                b0 = hi ? 0.0f : bOrog(52);
                b1 = hi ? 0.0f : bOrog(53);
            }
            v2f A = { a0, a1 };
            v2f B = { b0, b1 };
            acc = __builtin_amdgcn_wmma_f32_16x16x4_f32(
                false, A, false, B, (short)0, acc, false, false);
        }

        // Extract D[m, cls_ids[b]*4+f] + bias_high; store predicated only.
        #pragma unroll
        for (int j = 0; j < 8; ++j) {
            const float v = acc[j] + bias_high[bhBase[j] + p];
            if (match[j]) out[outBase[j] + p] = v;
        }
    }
}

extern "C" void kernel_launch(void* const* d_in, const int* in_sizes, int n_in,
                              void* d_out, int out_size, void* d_ws, size_t ws_size,
                              hipStream_t stream) {
    (void)in_sizes; (void)n_in; (void)d_ws; (void)ws_size; (void)out_size;
    const float* x         = (const float*)d_in[0];
    const float* orog      = (const float*)d_in[1];
    const float* wm        = (const float*)d_in[2];
    const float* bias_low  = (const float*)d_in[3];
    const float* bias_high = (const float*)d_in[4];
    const float* bias_orog = (const float*)d_in[5];
    const int*   cls_ids   = (const int*)d_in[6];
    const int*   nbi       = (const int*)d_in[7];
    float*       out       = (float*)d_out;

    dim3 grid(128);   // one block per high-res row
    dim3 block(256);  // 8 waves: 4 M-tiles x 2 p-halves
    rect_upsampler_orog_wmma<<<grid, block, 0, stream>>>(
        x, orog, wm, bias_low, bias_high, bias_orog, cls_ids, nbi, out);
}